// RGCN_2688649527602
// MI455X (gfx1250) — compile-verified
//
#include <hip/hip_runtime.h>

typedef __attribute__((ext_vector_type(2))) float v2f;
typedef __attribute__((ext_vector_type(8))) float v8f;

// ---------------------------------------------------------------------------
// Zero a float region (sums + cnt), float4 grid-stride.
// ---------------------------------------------------------------------------
__global__ void k_zero(float4* __restrict__ p, long long n4) {
  long long i = (long long)blockIdx.x * blockDim.x + threadIdx.x;
  long long stride = (long long)gridDim.x * blockDim.x;
  float4 z; z.x = 0.f; z.y = 0.f; z.z = 0.f; z.w = 0.f;
  for (; i < n4; i += stride) p[i] = z;
}

// ---------------------------------------------------------------------------
// Pack weights for WMMA B-fragment loads: wpack[col*1152 + k] where rows
// k<1024 come from wrel [1024,ncol] and k>=1024 from wroot [128,ncol].
// After packing, a lane's (k, k+1) B pair is one contiguous b64 load.
// ---------------------------------------------------------------------------
__global__ void k_pack(const float* __restrict__ wrel,
                       const float* __restrict__ wroot,
                       float* __restrict__ pack, int ncol) {
  int i = blockIdx.x * blockDim.x + threadIdx.x;
  int total = ncol * 1152;
  if (i < total) {
    int col = i / 1152;
    int k = i - col * 1152;
    pack[i] = (k < 1024) ? wrel[(size_t)k * ncol + col]
                         : wroot[(size_t)(k - 1024) * ncol + col];
  }
}

// ---------------------------------------------------------------------------
// Edge aggregation: one wave (32 lanes) per edge.
//   seg = dst*8 + edge_type ; sums[seg][0:128] += x[src][0:128] ; cnt[seg] += 1
// Lane l handles columns [4l, 4l+4): one float4 gather + 4 f32 atomics.
// ---------------------------------------------------------------------------
__global__ void k_edge_agg(const float* __restrict__ x,
                           const long long* __restrict__ ei,
                           const long long* __restrict__ et,
                           float* __restrict__ sums,
                           float* __restrict__ cnt,
                           int E) {
  const int lane = threadIdx.x & 31;
  long long wid = ((long long)blockIdx.x * blockDim.x + threadIdx.x) >> 5;
  long long nw = ((long long)gridDim.x * blockDim.x) >> 5;
  for (long long e = wid; e < E; e += nw) {
    long long src = ei[e];
    long long dst = ei[(long long)E + e];
    long long seg = dst * 8 + et[e];
    if (lane == 0) atomicAdd(&cnt[seg], 1.0f);
    const float4 v = *(const float4*)(x + src * 128 + lane * 4);
    float* sp = sums + seg * 128 + lane * 4;
    atomicAdd(sp + 0, v.x);
    atomicAdd(sp + 1, v.y);
    atomicAdd(sp + 2, v.z);
    atomicAdd(sp + 3, v.w);
  }
}

// ---------------------------------------------------------------------------
// Layer 1/2 GEMM: out[n,0:128] = relu( [means | x] @ [wrel ; wroot] + b )
//   K = 1152 = 8*128 (scaled relation sums) + 128 (root block).
//   Block: 64 nodes x 128 cols, 256 threads = 8 waves.
//   Wave w owns cols [16w,16w+16) and 4 row tiles (4 accumulators), so each
//   packed-B b64 load feeds 4 V_WMMA_F32_16X16X4_F32.
// ---------------------------------------------------------------------------
__global__ void __launch_bounds__(256) k_gemm128(
    const float* __restrict__ sums,   // [N, 1024]
    const float* __restrict__ cnt,    // [N, 8]
    const float* __restrict__ ax,     // [N, 128] (x or previous h)
    const float* __restrict__ wpack,  // [128, 1152] packed
    const float* __restrict__ bias,   // [128]
    float* __restrict__ out,          // [N, 128]
    int N) {
  __shared__ float As[64][132];  // 33.8 KB, stride keeps float4 alignment
  const int tid = threadIdx.x;
  const int lane = tid & 31;
  const int wave = tid >> 5;
  const int node0 = blockIdx.x * 64;
  const int rlo = lane & 15;
  const int khalf = (lane >> 4) << 1;       // 0 or 2
  const int col = wave * 16 + (lane & 15);  // output column
  const float* pk = wpack + (size_t)col * 1152;
  v8f acc0 = {}, acc1 = {}, acc2 = {}, acc3 = {};

  for (int c = 0; c < 9; ++c) {
    // ---- stage A chunk [64 x 128] into LDS (mean scaling folded in) ----
    for (int i = tid; i < 64 * 32; i += 256) {
      int rr = i >> 5;
      int d4 = (i & 31) << 2;
      int node = node0 + rr;
      float4 v; v.x = 0.f; v.y = 0.f; v.z = 0.f; v.w = 0.f;
      if (node < N) {
        if (c < 8) {
          float s = 1.0f / fmaxf(cnt[(size_t)node * 8 + c], 1.0f);
          float4 t = *(const float4*)(sums + (size_t)node * 1024 +
                                      (size_t)c * 128 + d4);
          v.x = t.x * s; v.y = t.y * s; v.z = t.z * s; v.w = t.w * s;
        } else {
          v = *(const float4*)(ax + (size_t)node * 128 + d4);
        }
      }
      *(float4*)&As[rr][d4] = v;
    }
    __syncthreads();

    const int kb = c * 128;
    for (int kk = 0; kk < 128; kk += 4) {
      const int kp = kk + khalf;
      v2f b = *(const v2f*)(pk + kb + kp);  // contiguous packed pair
      v2f a0, a1, a2, a3;
      a0.x = As[rlo][kp];      a0.y = As[rlo][kp + 1];
      a1.x = As[16 + rlo][kp]; a1.y = As[16 + rlo][kp + 1];
      a2.x = As[32 + rlo][kp]; a2.y = As[32 + rlo][kp + 1];
      a3.x = As[48 + rlo][kp]; a3.y = As[48 + rlo][kp + 1];
      acc0 = __builtin_amdgcn_wmma_f32_16x16x4_f32(false, a0, false, b,
                                                   (short)0, acc0, false, false);
      acc1 = __builtin_amdgcn_wmma_f32_16x16x4_f32(false, a1, false, b,
                                                   (short)0, acc1, false, false);
      acc2 = __builtin_amdgcn_wmma_f32_16x16x4_f32(false, a2, false, b,
                                                   (short)0, acc2, false, false);
      acc3 = __builtin_amdgcn_wmma_f32_16x16x4_f32(false, a3, false, b,
                                                   (short)0, acc3, false, false);
    }
    __syncthreads();
  }

  // ---- store: VGPR i of tile t holds C[M = 16t + i + 8*(lane>=16)][col] ----
  const int mhigh = (lane >> 4) << 3;
  const float bv = bias[col];
  for (int i = 0; i < 8; ++i) {
    int m = mhigh + i;
    float v;
    v = fmaxf(acc0[i] + bv, 0.0f);
    if (node0 + m < N)      out[(size_t)(node0 + m) * 128 + col] = v;
    v = fmaxf(acc1[i] + bv, 0.0f);
    if (node0 + 16 + m < N) out[(size_t)(node0 + 16 + m) * 128 + col] = v;
    v = fmaxf(acc2[i] + bv, 0.0f);
    if (node0 + 32 + m < N) out[(size_t)(node0 + 32 + m) * 128 + col] = v;
    v = fmaxf(acc3[i] + bv, 0.0f);
    if (node0 + 48 + m < N) out[(size_t)(node0 + 48 + m) * 128 + col] = v;
  }
}

// ---------------------------------------------------------------------------
// Layer 3: only the NROOT gathered rows are needed. Blocks of 1 wave,
// each wave: 16 root nodes x 16 output classes, K = 1152, no relu.
// ---------------------------------------------------------------------------
__global__ void __launch_bounds__(32) k_gemm_out(
    const float* __restrict__ sums,   // [N, 1024]
    const float* __restrict__ cnt,    // [N, 8]
    const float* __restrict__ ax,     // [N, 128] (h2)
    const float* __restrict__ wpack,  // [16, 1152] packed
    const float* __restrict__ bias,   // [16]
    const long long* __restrict__ root_idx,
    float* __restrict__ out) {        // [NROOT, 16]
  __shared__ int nid[16];
  const int lane = threadIdx.x & 31;
  if (lane < 16) nid[lane] = (int)root_idx[blockIdx.x * 16 + lane];
  __syncthreads();

  const int row = lane & 15;
  const int khalf = (lane >> 4) << 1;
  const int col = lane & 15;
  const int node = nid[row];
  const float* pk = wpack + (size_t)col * 1152;
  v8f acc = {};

  for (int kk = 0; kk < 1152; kk += 4) {
    const int kp = kk + khalf;
    const int c = kp >> 7;
    const int kc = kp & 127;
    v2f a;
    v2f b = *(const v2f*)(pk + kp);
    if (c < 8) {
      float s = 1.0f / fmaxf(cnt[(size_t)node * 8 + c], 1.0f);
      a.x = sums[(size_t)node * 1024 + kp] * s;
      a.y = sums[(size_t)node * 1024 + kp + 1] * s;
    } else {
      a.x = ax[(size_t)node * 128 + kc];
      a.y = ax[(size_t)node * 128 + kc + 1];
    }
    acc = __builtin_amdgcn_wmma_f32_16x16x4_f32(false, a, false, b, (short)0,
                                                acc, false, false);
  }

  const int mhigh = (lane >> 4) << 3;
  const float bv = bias[col];
  for (int i = 0; i < 8; ++i) {
    int m = mhigh + i;
    out[(size_t)(blockIdx.x * 16 + m) * 16 + col] = acc[i] + bv;
  }
}

// ---------------------------------------------------------------------------
extern "C" void kernel_launch(void* const* d_in, const int* in_sizes, int n_in,
                              void* d_out, int out_size, void* d_ws, size_t ws_size,
                              hipStream_t stream) {
  const float* x            = (const float*)d_in[0];
  const long long* ei       = (const long long*)d_in[1];
  const long long* et       = (const long long*)d_in[2];
  const long long* root_idx = (const long long*)d_in[3];
  const float* wrel1 = (const float*)d_in[4];
  const float* root1 = (const float*)d_in[5];
  const float* b1    = (const float*)d_in[6];
  const float* wrel2 = (const float*)d_in[7];
  const float* root2 = (const float*)d_in[8];
  const float* b2    = (const float*)d_in[9];
  const float* wrel3 = (const float*)d_in[10];
  const float* root3 = (const float*)d_in[11];
  const float* b3    = (const float*)d_in[12];

  const int N = in_sizes[0] / 128;
  const int E = in_sizes[2];

  // workspace (f32): sums [N*8*128] | cnt [N*8] | h1 [N*128] | h2 [N*128] | pack [128*1152]
  float* sums = (float*)d_ws;
  float* cnt  = sums + (size_t)N * 8 * 128;
  float* h1   = cnt + (size_t)N * 8;
  float* h2   = h1 + (size_t)N * 128;
  float* pack = h2 + (size_t)N * 128;

  const long long zn4 = ((long long)N * 8 * 128 + (long long)N * 8) / 4;
  const int zblocks = 2048, eblocks = 4096;
  const int gblocks = (N + 63) / 64;
  const int oblocks = in_sizes[3] / 16;
  const int p128 = (128 * 1152 + 255) / 256;
  const int p16  = (16 * 1152 + 255) / 256;

  // ---- layer 1 ----
  k_zero<<<zblocks, 256, 0, stream>>>((float4*)sums, zn4);
  k_pack<<<p128, 256, 0, stream>>>(wrel1, root1, pack, 128);
  k_edge_agg<<<eblocks, 256, 0, stream>>>(x, ei, et, sums, cnt, E);
  k_gemm128<<<gblocks, 256, 0, stream>>>(sums, cnt, x, pack, b1, h1, N);

  // ---- layer 2 ----
  k_zero<<<zblocks, 256, 0, stream>>>((float4*)sums, zn4);
  k_pack<<<p128, 256, 0, stream>>>(wrel2, root2, pack, 128);
  k_edge_agg<<<eblocks, 256, 0, stream>>>(h1, ei, et, sums, cnt, E);
  k_gemm128<<<gblocks, 256, 0, stream>>>(sums, cnt, h1, pack, b2, h2, N);

  // ---- layer 3 (only root rows) ----
  k_zero<<<zblocks, 256, 0, stream>>>((float4*)sums, zn4);
  k_pack<<<p16, 256, 0, stream>>>(wrel3, root3, pack, 16);
  k_edge_agg<<<eblocks, 256, 0, stream>>>(h2, ei, et, sums, cnt, E);
  k_gemm_out<<<oblocks, 32, 0, stream>>>(sums, cnt, h2, pack, b3,
                                         root_idx, (float*)d_out);
}